// HebbianLoss_54932631716110
// MI455X (gfx1250) — compile-verified
//
#include <hip/hip_runtime.h>
#include <hip/hip_bf16.h>
#include <math.h>

// ---- CDNA5 / gfx1250 types ----
typedef __attribute__((ext_vector_type(2))) float v2f;
typedef __attribute__((ext_vector_type(8))) float v8f;

// Problem constants (B=16, T=1024, I=1024)
#define HL_I      1024
#define HL_NROWS  16384        // B*T
#define HL_RTILES 1024         // HL_NROWS / 16

// LDS: S tile stored K-pair-transposed with XOR swizzle.
// Element (row m in 0..15, col j in 0..1023) lives at
//   sS[(j>>1)*16 + (m ^ ((j>>1)&15))]  component (j&1)
// -> A-fragment ds_load_b64 (16 lanes, fixed j2, m=0..15) is bank-conflict-free,
//    and since (j2&15) only depends on (k mod 8), all 256 K-steps reuse 8 base
//    addresses + constant DS immediate offsets (u*2048B <= 63488 < 64K).
#define SWZ(j2, m) (((j2) << 4) + ((m) ^ ((j2) & 15)))

__global__ __launch_bounds__(256)
void HebbianLoss_main_kernel(const float* __restrict__ S,
                             const float* __restrict__ W,
                             float* __restrict__ blockPartial) {
    __shared__ v2f sS[512 * 16];                  // exactly 64 KB

    const int t       = threadIdx.x;
    const int lane    = t & 31;
    const int wave    = t >> 5;                   // 0..7
    const int h       = lane >> 4;                // half-wave: K pair {0,1} vs {2,3}
    const int mA      = lane & 15;                // row (A) / col (B,C,D) within tile
    const int rowbase = blockIdx.x << 4;

    // ---- cooperative fill: 16 rows x 1024 floats -> LDS (coalesced 256B/wave) ----
    #pragma unroll 4
    for (int q = 0; q < 32; ++q) {
        int idx = t + (q << 8);                   // 0..8191 float2 slots
        int m   = idx >> 9;                       // 512 pairs per row
        int j2  = idx & 511;
        v2f v = *(const v2f*)(S + (size_t)(rowbase + m) * HL_I + (j2 << 1));
        sS[SWZ(j2, m)] = v;
    }

    // ---- hoist the 8 swizzled A-fragment base addresses (k = s mod 8) ----
    // k-step s: j2 = 2s + h (mod 16 class); address advances by 256 v2f per u.
    const v2f* aPtr[8];
    #pragma unroll
    for (int s = 0; s < 8; ++s)
        aPtr[s] = &sS[SWZ(2 * s + h, mA)];

    __syncthreads();

    // ---- fused GEMM (WS = S_tile * W^T) + elementwise dot ----
    // Each wave owns 8 i-tiles, processed as 4 pairs: two accumulators share
    // every A fragment (halves DS traffic, two independent WMMA dep-chains).
    float acc[8] = {0.f, 0.f, 0.f, 0.f, 0.f, 0.f, 0.f, 0.f};

    for (int g = 0; g < 4; ++g) {
        const int i0a = (wave + (g << 4)) << 4;   // itile = wave + 16g
        const int i0b = i0a + 128;                // itile + 8
        const float* wrow0 = W + (size_t)(i0a + mA) * HL_I + (h << 1);
        const float* wrow1 = W + (size_t)(i0b + mA) * HL_I + (h << 1);

        v8f c0 = {0.f, 0.f, 0.f, 0.f, 0.f, 0.f, 0.f, 0.f};
        v8f c1 = {0.f, 0.f, 0.f, 0.f, 0.f, 0.f, 0.f, 0.f};

        #pragma unroll 2
        for (int u = 0; u < 32; ++u) {
            // stream 2KB ahead in each lane's W row (global_prefetch_b8)
            __builtin_prefetch(wrow0 + (u << 5) + 512, 0, 0);
            __builtin_prefetch(wrow1 + (u << 5) + 512, 0, 0);
            #pragma unroll
            for (int s = 0; s < 8; ++s) {
                // A: S[m][j0+2h..+1], j0 = 4*(s + 8u)
                v2f a  = aPtr[s][u << 8];         // base + u*2048B (DS immediate)
                v2f b0 = *(const v2f*)(wrow0 + (u << 5) + (s << 2));
                v2f b1 = *(const v2f*)(wrow1 + (u << 5) + (s << 2));
                c0 = __builtin_amdgcn_wmma_f32_16x16x4_f32(
                         false, a, false, b0, (short)0, c0, false, false);
                c1 = __builtin_amdgcn_wmma_f32_16x16x4_f32(
                         false, a, false, b1, (short)0, c1, false, false);
            }
        }

        // elementwise: acc_row += WS[M][N] * S[M][i+N] for both i-tiles
        const int na = i0a + mA;                  // N = lane&15
        const int nb = i0b + mA;
        const int j2a = na >> 1, loa = na & 1;
        const int j2b = nb >> 1, lob = nb & 1;
        #pragma unroll
        for (int r = 0; r < 8; ++r) {
            const int M = r + (h << 3);           // D layout: VGPR r -> M = r + 8*(lane/16)
            v2f sva = sS[SWZ(j2a, M)];
            v2f svb = sS[SWZ(j2b, M)];
            acc[r] += c0[r] * (loa ? sva.y : sva.x);
            acc[r] += c1[r] * (lob ? svb.y : svb.x);
        }
    }

    // ---- reduce acc over the 16 lanes sharing each row ----
    #pragma unroll
    for (int r = 0; r < 8; ++r) {
        float v = acc[r];
        v += __shfl_xor(v, 1, 32);
        v += __shfl_xor(v, 2, 32);
        v += __shfl_xor(v, 4, 32);
        v += __shfl_xor(v, 8, 32);
        acc[r] = v;                               // lanes 0 and 16 hold group sums
    }

    __syncthreads();                              // done reading sS; reuse as scratch
    float* rowBuf = (float*)sS;                   // [8 waves][16 rows]
    if ((lane & 15) == 0) {
        #pragma unroll
        for (int r = 0; r < 8; ++r)
            rowBuf[(wave << 4) + (h << 3) + r] = acc[r];
    }
    __syncthreads();

    // deterministic per-block combine: sum waves in fixed order, |.| per row, sum
    if (t == 0) {
        float bs = 0.f;
        #pragma unroll
        for (int m = 0; m < 16; ++m) {
            float rs = 0.f;
            #pragma unroll
            for (int w2 = 0; w2 < 8; ++w2) rs += rowBuf[(w2 << 4) + m];
            bs += fabsf(rs);
        }
        blockPartial[blockIdx.x] = bs;
    }
}

__global__ __launch_bounds__(256)
void HebbianLoss_reduce_kernel(const float* __restrict__ partials,
                               float* __restrict__ out,
                               int n, float invN) {
    __shared__ float sm[256];
    float s = 0.f;
    for (int i = threadIdx.x; i < n; i += 256) s += partials[i];
    sm[threadIdx.x] = s;
    __syncthreads();
    #pragma unroll
    for (int off = 128; off > 0; off >>= 1) {
        if ((int)threadIdx.x < off) sm[threadIdx.x] += sm[threadIdx.x + off];
        __syncthreads();
    }
    if (threadIdx.x == 0) out[0] = sm[0] * invN;
}

extern "C" void kernel_launch(void* const* d_in, const int* in_sizes, int n_in,
                              void* d_out, int out_size, void* d_ws, size_t ws_size,
                              hipStream_t stream) {
    const float* S = (const float*)d_in[0];       // state  [16,1024,1024] fp32
    const float* W = (const float*)d_in[1];       // weights[1024,1024]    fp32
    float* partial = (float*)d_ws;                // 1024 floats scratch
    float* out     = (float*)d_out;               // scalar

    HebbianLoss_main_kernel<<<dim3(HL_RTILES), dim3(256), 0, stream>>>(S, W, partial);
    HebbianLoss_reduce_kernel<<<dim3(1), dim3(256), 0, stream>>>(
        partial, out, HL_RTILES, 1.0f / (float)HL_NROWS);
}